// NodeAndHyperedges_79998060855736
// MI455X (gfx1250) — compile-verified
//
#include <hip/hip_runtime.h>

typedef __attribute__((ext_vector_type(2))) float v2f;
typedef __attribute__((ext_vector_type(8))) float v8f;

#define N_NODES 50000
#define M_HEDGE 10000
#define E_INC   600000
#define HDIM    128

__device__ __forceinline__ float leakyf(float v, float s) { return v > 0.f ? v : s * v; }

// monotone order-preserving float<->uint encoding (for atomic max/min on floats)
__device__ __forceinline__ unsigned f32_ord(float f) {
    unsigned u = __float_as_uint(f);
    return (u & 0x80000000u) ? ~u : (u | 0x80000000u);
}
__device__ __forceinline__ float ord_f32(unsigned u) {
    unsigned b = (u & 0x80000000u) ? (u ^ 0x80000000u) : ~u;
    return __uint_as_float(b);
}

// ---------------------------------------------------------------------------
// GEMM: out[rows x NT*16] = act( (A (+Aadd)) @ W + bias )
// fp32 WMMA 16x16x4; W staged into LDS in 128-row chunks using the CDNA5
// async global->LDS copy path (ASYNCcnt). 256 thr = 8 waves; each wave owns a
// 16-row stripe and all NT column tiles, reading B fragments from LDS.
// ---------------------------------------------------------------------------
template <int NT, bool HASADD>
__global__ void gemm_wmma_f32(const float* __restrict__ A, const float* __restrict__ Aadd,
                              const float* __restrict__ W, const float* __restrict__ bias,
                              float* __restrict__ out, int rows, int K, int act) {
    const int NCOLS  = NT * 16;
    const int KCHUNK = 128;
    __shared__ float shW[KCHUNK * NCOLS];   // 64KB (NT=8), 32KB (NT=4)

    int tile = blockIdx.x * (blockDim.x >> 5) + (threadIdx.x >> 5);
    bool active = (tile * 16 < rows);       // wave-uniform
    int lane = threadIdx.x & 31;
    int half = lane >> 4;
    int l16  = lane & 15;
    int rowA = tile * 16 + l16;

    v8f acc[NT];
#pragma unroll
    for (int nt = 0; nt < NT; ++nt)
#pragma unroll
        for (int g = 0; g < 8; ++g) acc[nt][g] = 0.f;

    const float* arow  = A + (size_t)rowA * K;
    const float* arow2 = HASADD ? (Aadd + (size_t)rowA * K) : nullptr;
    unsigned ldsBase = (unsigned)(uintptr_t)(&shW[0]);   // low 32 bits = LDS offset

    for (int kc = 0; kc < K; kc += KCHUNK) {
        // ---- stage W chunk into LDS with async copies (16B per lane-op) ----
        const float* wsrc = W + (size_t)kc * NCOLS;
        const int nvec = (KCHUNK * NCOLS) / 4;           // # of float4s
        for (int t = threadIdx.x; t < nvec; t += blockDim.x) {
            unsigned laddr = ldsBase + (unsigned)t * 16u;
            unsigned long long ga = (unsigned long long)(uintptr_t)(wsrc + t * 4);
            asm volatile("global_load_async_to_lds_b128 %0, %1, off"
                         :: "v"(laddr), "v"(ga) : "memory");
        }
        asm volatile("s_wait_asynccnt 0x0" ::: "memory");
        __syncthreads();

        if (active) {
            for (int k = 0; k < KCHUNK; k += 4) {
                int ka = k + 2 * half;
                float a0 = arow[kc + ka];
                float a1 = arow[kc + ka + 1];
                if (HASADD) { a0 += arow2[kc + ka]; a1 += arow2[kc + ka + 1]; }
                v2f af; af[0] = a0; af[1] = a1;
                const float* w0 = &shW[ka * NCOLS];
                const float* w1 = w0 + NCOLS;
#pragma unroll
                for (int nt = 0; nt < NT; ++nt) {
                    int col = nt * 16 + l16;
                    v2f bf; bf[0] = w0[col]; bf[1] = w1[col];
                    acc[nt] = __builtin_amdgcn_wmma_f32_16x16x4_f32(
                        false, af, false, bf, (short)0, acc[nt], false, false);
                }
            }
        }
        __syncthreads();   // all waves: protect shW before next stage
    }

    if (active) {
#pragma unroll
        for (int nt = 0; nt < NT; ++nt) {
            int col = nt * 16 + l16;
            float bv = bias ? bias[col] : 0.f;
#pragma unroll
            for (int g = 0; g < 8; ++g) {
                int row = tile * 16 + g + 8 * half;
                float v = acc[nt][g] + bv;
                if (act) v = leakyf(v, 0.01f);
                out[(size_t)row * NCOLS + col] = v;
            }
        }
    }
}

// ---------------------------------------------------------------------------
// Edge / segment kernels
// ---------------------------------------------------------------------------
__global__ void count_kernel(const int* __restrict__ ni, const int* __restrict__ hi,
                             float* __restrict__ cnt_v, float* __restrict__ cnt_e) {
    int e = blockIdx.x * blockDim.x + threadIdx.x;
    if (e >= E_INC) return;
    atomicAdd(&cnt_e[hi[e]], 1.f);
    atomicAdd(&cnt_v[ni[e]], 1.f);
}

// one wave per edge; lane handles 4 features
__global__ void edge_scatter_add_kernel(const float* __restrict__ src, const int* __restrict__ sidx,
                                        const int* __restrict__ didx, float* __restrict__ dst) {
    int e = blockIdx.x * (blockDim.x >> 5) + (threadIdx.x >> 5);
    if (e >= E_INC) return;
    int lane = threadIdx.x & 31;
    const float4* sp = (const float4*)(src + (size_t)sidx[e] * HDIM);
    float4 v = sp[lane];
    float* dp = dst + (size_t)didx[e] * HDIM + lane * 4;
    atomicAdd(dp + 0, v.x); atomicAdd(dp + 1, v.y);
    atomicAdd(dp + 2, v.z); atomicAdd(dp + 3, v.w);
}

__global__ void rowdiv_kernel(float* __restrict__ x, const float* __restrict__ cnt, int rows) {
    int i = blockIdx.x * blockDim.x + threadIdx.x;
    if (i >= rows * HDIM) return;
    x[i] /= fmaxf(cnt[i / HDIM], 1.f);
}

__global__ void edge_dot_kernel(const float* __restrict__ q, const float* __restrict__ kk,
                                const int* __restrict__ ni, const int* __restrict__ hi,
                                float* __restrict__ s, unsigned* __restrict__ smax_u) {
    int e = blockIdx.x * (blockDim.x >> 5) + (threadIdx.x >> 5);
    if (e >= E_INC) return;
    int lane = threadIdx.x & 31;
    int n = ni[e], h = hi[e];
    float4 a = ((const float4*)(q + (size_t)n * HDIM))[lane];
    float4 b = ((const float4*)(kk + (size_t)h * HDIM))[lane];
    float d = a.x * b.x + a.y * b.y + a.z * b.z + a.w * b.w;
    for (int off = 16; off; off >>= 1) d += __shfl_xor(d, off, 32);
    if (lane == 0) {
        float sv = leakyf(d * 0.08838834764831845f, 0.2f);  // 1/sqrt(128), ATT_SLOPE
        s[e] = sv;
        atomicMax(&smax_u[n], f32_ord(sv));
    }
}

__global__ void edge_exp_kernel(const float* __restrict__ s, const unsigned* __restrict__ smax_u,
                                const int* __restrict__ ni, float* __restrict__ eexp,
                                float* __restrict__ denom) {
    int e = blockIdx.x * blockDim.x + threadIdx.x;
    if (e >= E_INC) return;
    int n = ni[e];
    float ee = expf(s[e] - ord_f32(smax_u[n]));
    eexp[e] = ee;
    atomicAdd(&denom[n], ee);
}

__global__ void edge_av_kernel(const float* __restrict__ eexp, const float* __restrict__ denom,
                               const float* __restrict__ vv, const int* __restrict__ ni,
                               const int* __restrict__ hi, float* __restrict__ hatt) {
    int e = blockIdx.x * (blockDim.x >> 5) + (threadIdx.x >> 5);
    if (e >= E_INC) return;
    int lane = threadIdx.x & 31;
    int n = ni[e], h = hi[e];
    float alpha = eexp[e] / fmaxf(denom[n], 1e-16f);
    float4 v4 = ((const float4*)(vv + (size_t)h * HDIM))[lane];
    float* dp = hatt + (size_t)n * HDIM + lane * 4;
    atomicAdd(dp + 0, alpha * v4.x); atomicAdd(dp + 1, alpha * v4.y);
    atomicAdd(dp + 2, alpha * v4.z); atomicAdd(dp + 3, alpha * v4.w);
}

__global__ void edge_min_kernel(const float* __restrict__ hatt, const int* __restrict__ ni,
                                const int* __restrict__ hi, unsigned* __restrict__ agg_u) {
    int e = blockIdx.x * (blockDim.x >> 5) + (threadIdx.x >> 5);
    if (e >= E_INC) return;
    int lane = threadIdx.x & 31;
    float4 v = ((const float4*)(hatt + (size_t)ni[e] * HDIM))[lane];
    unsigned* dp = agg_u + (size_t)hi[e] * HDIM + lane * 4;
    atomicMin(dp + 0, f32_ord(v.x)); atomicMin(dp + 1, f32_ord(v.y));
    atomicMin(dp + 2, f32_ord(v.z)); atomicMin(dp + 3, f32_ord(v.w));
}

// z1[m, 0:128] = (cnt_e>0.5 ? decode(agg) : 0), z1[m, 128:256] = xe[m]
__global__ void build_z1_kernel(const unsigned* __restrict__ agg_u, const float* __restrict__ cnt_e,
                                const float* __restrict__ xe, float* __restrict__ z1) {
    int i = blockIdx.x * blockDim.x + threadIdx.x;
    if (i >= M_HEDGE * HDIM) return;
    int m = i / HDIM, f = i % HDIM;
    float av = (cnt_e[m] > 0.5f) ? ord_f32(agg_u[i]) : 0.f;
    z1[(size_t)m * 256 + f] = av;
    z1[(size_t)m * 256 + 128 + f] = xe[i];
}

// ---------------------------------------------------------------------------
// GraphNorm: per-column mean + rsqrt(var); var = E[x^2] - 2a*mu^2 + a^2*mu^2
// ---------------------------------------------------------------------------
__global__ void colstats_kernel(const float* __restrict__ X, int rows, int cols,
                                const float* __restrict__ avec,
                                float* __restrict__ mu_out, float* __restrict__ rinv_out) {
    int c = blockIdx.x;
    float s1 = 0.f, s2 = 0.f;
    for (int r = threadIdx.x; r < rows; r += blockDim.x) {
        float x = X[(size_t)r * cols + c];
        s1 += x; s2 += x * x;
    }
    __shared__ float sh1[256], sh2[256];
    sh1[threadIdx.x] = s1; sh2[threadIdx.x] = s2;
    __syncthreads();
    for (int st = blockDim.x >> 1; st > 0; st >>= 1) {
        if (threadIdx.x < st) {
            sh1[threadIdx.x] += sh1[threadIdx.x + st];
            sh2[threadIdx.x] += sh2[threadIdx.x + st];
        }
        __syncthreads();
    }
    if (threadIdx.x == 0) {
        float inv = 1.f / (float)rows;
        float mu = sh1[0] * inv;
        float a = avec[c];
        float var = sh2[0] * inv - 2.f * a * mu * mu + a * a * mu * mu;
        mu_out[c] = mu;
        rinv_out[c] = rsqrtf(var + 1e-5f);
    }
}

__global__ void apply_gn_kernel(float* __restrict__ x, int rows, int cols,
                                const float* __restrict__ w, const float* __restrict__ b,
                                const float* __restrict__ a, const float* __restrict__ mu,
                                const float* __restrict__ rinv, int act) {
    int i = blockIdx.x * blockDim.x + threadIdx.x;
    if (i >= rows * cols) return;
    int c = i % cols;
    float y = w[c] * (x[i] - a[c] * mu[c]) * rinv[c] + b[c];
    if (act) y = leakyf(y, 0.01f);
    x[i] = y;
}

// ---------------------------------------------------------------------------
extern "C" void kernel_launch(void* const* d_in, const int* in_sizes, int n_in,
                              void* d_out, int out_size, void* d_ws, size_t ws_size,
                              hipStream_t stream) {
    const float* x        = (const float*)d_in[0];
    const float* x_e      = (const float*)d_in[2];
    const int*   node_idx = (const int*)d_in[3];
    const int*   hedge_idx= (const int*)d_in[4];
    const float* her_W1 = (const float*)d_in[5];
    const float* her_b1 = (const float*)d_in[6];
    const float* her_W2 = (const float*)d_in[7];
    const float* her_b2 = (const float*)d_in[8];
    const float* sr_W1  = (const float*)d_in[9];
    const float* sr_b1  = (const float*)d_in[10];
    const float* sr_W2  = (const float*)d_in[11];
    const float* sr_b2  = (const float*)d_in[12];
    const float* att_Wq = (const float*)d_in[13];
    const float* att_Wk = (const float*)d_in[14];
    const float* att_Wv = (const float*)d_in[15];
    const float* ef_W   = (const float*)d_in[16];
    const float* ef_b   = (const float*)d_in[17];
    const float* gn1_w  = (const float*)d_in[18];
    const float* gn1_b  = (const float*)d_in[19];
    const float* gn1_a  = (const float*)d_in[20];
    const float* gn2_w  = (const float*)d_in[21];
    const float* gn2_b  = (const float*)d_in[22];
    const float* gn2_a  = (const float*)d_in[23];
    const float* cls_W1 = (const float*)d_in[24];
    const float* cls_b1 = (const float*)d_in[25];
    const float* cls_W2 = (const float*)d_in[26];
    const float* cls_b2 = (const float*)d_in[27];

    // workspace layout (fp32 elements)
    float* ws = (float*)d_ws;
    size_t o = 0;
    float* t1   = ws + o; o += (size_t)M_HEDGE * HDIM;
    float* xe   = ws + o; o += (size_t)M_HEDGE * HDIM;
    float* h1   = ws + o; o += (size_t)N_NODES * HDIM;
    float* m_e  = ws + o; o += (size_t)M_HEDGE * HDIM;
    float* m_v  = ws + o; o += (size_t)N_NODES * HDIM;
    float* h2   = ws + o; o += (size_t)N_NODES * HDIM;
    float* q    = ws + o; o += (size_t)N_NODES * HDIM;
    float* kk   = ws + o; o += (size_t)M_HEDGE * HDIM;
    float* vv   = ws + o; o += (size_t)M_HEDGE * HDIM;
    float* s    = ws + o; o += (size_t)E_INC;
    float* eexp = ws + o; o += (size_t)E_INC;
    unsigned* smax_u = (unsigned*)(ws + o); o += (size_t)N_NODES;
    float* denom= ws + o; o += (size_t)N_NODES;
    float* cnt_e= ws + o; o += (size_t)M_HEDGE;
    float* cnt_v= ws + o; o += (size_t)N_NODES;
    float* hatt = ws + o; o += (size_t)N_NODES * HDIM;
    unsigned* agg_u = (unsigned*)(ws + o); o += (size_t)M_HEDGE * HDIM;
    float* z1   = ws + o; o += (size_t)M_HEDGE * 256;
    float* z2   = ws + o; o += (size_t)M_HEDGE * HDIM;
    float* z3   = ws + o; o += (size_t)M_HEDGE * HDIM;
    float* mu1  = ws + o; o += 256;
    float* ri1  = ws + o; o += 256;
    float* mu2  = ws + o; o += 128;
    float* ri2  = ws + o; o += 128;

    // clear accumulators (re-done every call: graph replays must be deterministic)
    hipMemsetAsync(m_e,   0,    (size_t)M_HEDGE * HDIM * 4, stream);
    hipMemsetAsync(m_v,   0,    (size_t)N_NODES * HDIM * 4, stream);
    hipMemsetAsync(hatt,  0,    (size_t)N_NODES * HDIM * 4, stream);
    hipMemsetAsync(denom, 0,    (size_t)N_NODES * 4, stream);
    hipMemsetAsync(cnt_e, 0,    (size_t)M_HEDGE * 4, stream);
    hipMemsetAsync(cnt_v, 0,    (size_t)N_NODES * 4, stream);
    hipMemsetAsync(smax_u,0,    (size_t)N_NODES * 4, stream);          // ordered-min value
    hipMemsetAsync(agg_u, 0xFF, (size_t)M_HEDGE * HDIM * 4, stream);   // ordered +max

    const int EWB = (E_INC + 7) / 8;          // wave-per-edge kernels, 256 thr = 8 waves
    const int ETB = (E_INC + 255) / 256;      // thread-per-edge kernels

    auto gemm8 = [&](const float* A, const float* W, const float* b,
                     float* out, int rows, int K, int act) {
        int blocks = (rows / 16 + 7) / 8;
        hipLaunchKernelGGL((gemm_wmma_f32<8, false>), dim3(blocks), dim3(256), 0, stream,
                           A, (const float*)nullptr, W, b, out, rows, K, act);
    };

    // --- hyperedge refiner: xe = act(x_e@W1+b1)@W2+b2
    gemm8(x_e, her_W1, her_b1, t1, M_HEDGE, HDIM, 1);
    gemm8(t1, her_W2, her_b2, xe, M_HEDGE, HDIM, 0);

    // --- node refiner MLP + one mean message-passing round
    gemm8(x, sr_W1, sr_b1, h1, N_NODES, HDIM, 1);
    count_kernel<<<ETB, 256, 0, stream>>>(node_idx, hedge_idx, cnt_v, cnt_e);
    edge_scatter_add_kernel<<<EWB, 256, 0, stream>>>(h1, node_idx, hedge_idx, m_e);
    rowdiv_kernel<<<(M_HEDGE * HDIM + 255) / 256, 256, 0, stream>>>(m_e, cnt_e, M_HEDGE);
    edge_scatter_add_kernel<<<EWB, 256, 0, stream>>>(m_e, hedge_idx, node_idx, m_v);
    rowdiv_kernel<<<(N_NODES * HDIM + 255) / 256, 256, 0, stream>>>(m_v, cnt_v, N_NODES);
    {   // h2 = act((h1+m_v)@sr_W2+sr_b2)
        int blocks = (N_NODES / 16 + 7) / 8;
        hipLaunchKernelGGL((gemm_wmma_f32<8, true>), dim3(blocks), dim3(256), 0, stream,
                           h1, m_v, sr_W2, sr_b2, h2, N_NODES, HDIM, 1);
    }

    // --- GAT-style attention node<-hyperedges
    gemm8(h2, att_Wq, nullptr, q,  N_NODES, HDIM, 0);
    gemm8(xe, att_Wk, nullptr, kk, M_HEDGE, HDIM, 0);
    gemm8(xe, att_Wv, nullptr, vv, M_HEDGE, HDIM, 0);
    edge_dot_kernel<<<EWB, 256, 0, stream>>>(q, kk, node_idx, hedge_idx, s, smax_u);
    edge_exp_kernel<<<ETB, 256, 0, stream>>>(s, smax_u, node_idx, eexp, denom);
    edge_av_kernel<<<EWB, 256, 0, stream>>>(eexp, denom, vv, node_idx, hedge_idx, hatt);

    // --- min-aggregation into hyperedges, concat with xe
    edge_min_kernel<<<EWB, 256, 0, stream>>>(hatt, node_idx, hedge_idx, agg_u);
    build_z1_kernel<<<(M_HEDGE * HDIM + 255) / 256, 256, 0, stream>>>(agg_u, cnt_e, xe, z1);

    // --- edge fusion: GN1 -> Linear(256,128)+act -> GN2 -> act
    colstats_kernel<<<256, 256, 0, stream>>>(z1, M_HEDGE, 256, gn1_a, mu1, ri1);
    apply_gn_kernel<<<(M_HEDGE * 256 + 255) / 256, 256, 0, stream>>>(
        z1, M_HEDGE, 256, gn1_w, gn1_b, gn1_a, mu1, ri1, 0);
    gemm8(z1, ef_W, ef_b, z2, M_HEDGE, 256, 1);
    colstats_kernel<<<128, 256, 0, stream>>>(z2, M_HEDGE, 128, gn2_a, mu2, ri2);
    apply_gn_kernel<<<(M_HEDGE * HDIM + 255) / 256, 256, 0, stream>>>(
        z2, M_HEDGE, 128, gn2_w, gn2_b, gn2_a, mu2, ri2, 1);

    // --- classifier
    gemm8(z2, cls_W1, cls_b1, z3, M_HEDGE, HDIM, 1);
    {
        int blocks = (M_HEDGE / 16 + 7) / 8;
        hipLaunchKernelGGL((gemm_wmma_f32<4, false>), dim3(blocks), dim3(256), 0, stream,
                           z3, (const float*)nullptr, cls_W2, cls_b2,
                           (float*)d_out, M_HEDGE, HDIM, 0);
    }
}